// image_to_word_tf_85839216378366
// MI455X (gfx1250) — compile-verified
//
#include <hip/hip_runtime.h>

// Problem constants (from reference)
#define BB 32
#define HH 512
#define VV 5000
#define TT 38
#define G3H 1536          // 3*H
#define NSTEP 703         // sum_{i=1..37} i  (total generated states)
#define SDIM 704          // output s-dimension = 1 + NSTEP
#define NWG 16            // workgroups in recurrent kernel
#define JCOLS 32          // hidden columns per WG
#define GCOLS 96          // gate columns per WG (3*JCOLS)
#define HLS 516           // padded hl row stride (bank-conflict-free A frags)
#define PSTR 224          // Wt pair-row stride (floats): 2*GCOLS + 32 -> half-waves hit disjoint banks
#define KBLK 32
#define AS_S 36
#define PB 160            // Bs pair-row stride (floats): 2*64 + 32

typedef __attribute__((ext_vector_type(2))) float v2f;
typedef __attribute__((ext_vector_type(8))) float v8f;

__device__ __forceinline__ float sigmoidf_(float x) { return 1.0f / (1.0f + __expf(-x)); }

// ---------------------------------------------------------------------------
// Kernel 0: reset grid-sync counter, seed h0 = img_feat, write one-hot s=0 col
// ---------------------------------------------------------------------------
__global__ void init_kernel(const float* __restrict__ img_feat,
                            float* __restrict__ hbuf,
                            float* __restrict__ out,
                            unsigned* __restrict__ syncp) {
  int idx = blockIdx.x * blockDim.x + threadIdx.x;
  if (idx == 0) *syncp = 0u;
  if (idx < BB * HH) hbuf[idx] = img_feat[idx];
  if (idx < BB * VV) {
    int b = idx / VV, v = idx % VV;
    out[((size_t)b * VV + v) * SDIM] = (v == 0) ? 1.0f : 0.0f;  // SOS one-hot
  }
}

// ---------------------------------------------------------------------------
// Kernel 1: Gi[t][b][j] = embed[tok(t,b)] . w_ih[j] + b_ih[j]   (37x32x1536)
// ---------------------------------------------------------------------------
__global__ void gi_kernel(const int* __restrict__ captions,
                          const float* __restrict__ embed,
                          const float* __restrict__ w_ih,
                          const float* __restrict__ b_ih,
                          float* __restrict__ Gi) {
  int j = blockIdx.x * blockDim.x + threadIdx.x;  // 0..1535
  int b = blockIdx.y;
  int t = blockIdx.z;                             // 0..36
  int tok = (t == 0) ? 0 : captions[b * TT + t];  // E[0]=SOS, E[t]=captions[b,t]
  const float4* e = (const float4*)(embed + (size_t)tok * HH);
  const float4* w = (const float4*)(w_ih + (size_t)j * HH);
  float acc = 0.f;
#pragma unroll 4
  for (int k = 0; k < HH / 4; ++k) {
    float4 ev = e[k], wv = w[k];
    acc += ev.x * wv.x + ev.y * wv.y + ev.z * wv.z + ev.w * wv.w;
  }
  Gi[(size_t)t * BB * G3H + (size_t)b * G3H + j] = acc + b_ih[j];
}

// ---------------------------------------------------------------------------
// Kernel 2: persistent recurrent kernel. 16 WGs x 384 threads (12 waves).
// Each WG owns 32 hidden columns -> 96 gate columns, w_hh slice LDS-resident
// in k-paired layout (one ds_load_b64 per B fragment, conflict-free banks).
// 703 sequential GRU steps with device-wide arrive/spin barrier per step.
// ---------------------------------------------------------------------------
__global__ void gru_kernel(const float* __restrict__ Gi,
                           const float* __restrict__ w_hh,
                           const float* __restrict__ b_hh,
                           float* __restrict__ hbuf,
                           float* __restrict__ st,
                           unsigned* __restrict__ syncp,
                           float* __restrict__ out_hidden) {
  extern __shared__ float smem[];
  float* hl  = smem;                     // BB*HLS        : full h, padded rows
  float* Wt  = smem + BB * HLS;          // (HH/2)*PSTR   : Wt[k/2][n][2] = w_hh[grow(n)][k]
  float* ghl = Wt + (HH / 2) * PSTR;     // BB*GCOLS      : gh spill for gate math

  const int tid = threadIdx.x;
  const int j0 = blockIdx.x * JCOLS;

  // One-time: stage weight slice, k-paired: Wt[(k>>1)*PSTR + 2n + (k&1)]
  for (int idx = tid; idx < GCOLS * HH; idx += blockDim.x) {
    int n = idx / HH, k = idx % HH;
    int gate = n / JCOLS, jj = n % JCOLS;
    int grow = gate * HH + j0 + jj;
    Wt[(k >> 1) * PSTR + 2 * n + (k & 1)] = w_hh[(size_t)grow * HH + k];
  }

  const int wave = tid >> 5, lane = tid & 31;
  const int row = lane & 15, hi = lane >> 4;   // hi selects K pair per f32 WMMA layout
  const int mt = wave & 1;                     // 2 M tiles (B=32)
  const int nt = wave >> 1;                    // 6 N tiles (96 gate cols)
  const int mrow = mt * 16 + row;
  const int ncol = nt * 16 + row;

  // Initial h stage (fresh from init_kernel; kernel-boundary coherence)
  {
    const float4* src = (const float4*)hbuf;
    for (int idx = tid; idx < BB * HH / 4; idx += blockDim.x) {
      int b = idx >> 7, k4 = idx & 127;
      *(float4*)&hl[b * HLS + 4 * k4] = src[idx];
    }
  }
  __syncthreads();

  int u = 0;
  for (int i = 0; i < TT - 1; ++i) {
    for (int t = 0; t <= i; ++t, ++u) {
      // C(32x96) = h(32x512) @ Wslice^T ; two interleaved accumulator chains
      v8f acc0 = {}, acc1 = {};
      for (int k0 = 0; k0 < HH; k0 += 8) {
        v2f a0 = *(const v2f*)&hl[mrow * HLS + k0 + 2 * hi];
        v2f b0 = *(const v2f*)&Wt[((k0 >> 1) + hi) * PSTR + 2 * ncol];
        acc0 = __builtin_amdgcn_wmma_f32_16x16x4_f32(false, a0, false, b0,
                                                     (short)0, acc0, false, false);
        v2f a1 = *(const v2f*)&hl[mrow * HLS + k0 + 4 + 2 * hi];
        v2f b1 = *(const v2f*)&Wt[(((k0 + 4) >> 1) + hi) * PSTR + 2 * ncol];
        acc1 = __builtin_amdgcn_wmma_f32_16x16x4_f32(false, a1, false, b1,
                                                     (short)0, acc1, false, false);
      }
      v8f acc = acc0 + acc1;
#pragma unroll
      for (int r = 0; r < 8; ++r)
        ghl[(mt * 16 + hi * 8 + r) * GCOLS + ncol] = acc[r];
      __syncthreads();

      // Gate math on 32x32 owned slice; write new h + state archive
      const float* gib = Gi + (size_t)t * BB * G3H;
      for (int idx = tid; idx < BB * JCOLS; idx += blockDim.x) {
        int b = idx >> 5, jj = idx & (JCOLS - 1);
        int jg = j0 + jj;
        float gr  = ghl[b * GCOLS + jj]             + b_hh[jg];
        float gz  = ghl[b * GCOLS + JCOLS + jj]     + b_hh[HH + jg];
        float gn  = ghl[b * GCOLS + 2 * JCOLS + jj] + b_hh[2 * HH + jg];
        float ir  = gib[b * G3H + jg];
        float iz  = gib[b * G3H + HH + jg];
        float inn = gib[b * G3H + 2 * HH + jg];
        float rg = sigmoidf_(ir + gr);
        float zg = sigmoidf_(iz + gz);
        float ng = tanhf(inn + rg * gn);
        float hold = hl[b * HLS + jg];
        float hnew = (1.0f - zg) * ng + zg * hold;
        hbuf[b * HH + jg] = hnew;
        st[(size_t)u * BB * HH + b * HH + jg] = hnew;
      }
      __threadfence();       // release h writes device-wide
      __syncthreads();       // all lanes' stores fenced before arrive

      if (tid == 0) {
        atomicAdd(syncp, 1u);
        unsigned target = (unsigned)(NWG * (u + 1));
        while (__hip_atomic_load(syncp, __ATOMIC_ACQUIRE, __HIP_MEMORY_SCOPE_AGENT) < target) {
          __builtin_amdgcn_s_sleep(1);
        }
      }
      __syncthreads();

      // Drop stale WGP$ lines, then restage full h with plain vector loads
      asm volatile("global_inv scope:SCOPE_DEV\n\t"
                   "s_wait_loadcnt 0x0" ::: "memory");
      {
        const float4* src = (const float4*)hbuf;
        for (int idx = tid; idx < BB * HH / 4; idx += blockDim.x) {
          int b = idx >> 7, k4 = idx & 127;
          *(float4*)&hl[b * HLS + 4 * k4] = src[idx];
        }
      }
      __syncthreads();
    }
  }

  // Final hidden -> output tail (hl holds the fully synced final h)
  if (blockIdx.x == 0) {
    for (int idx = tid; idx < BB * HH; idx += blockDim.x) {
      int b = idx >> 9, k = idx & (HH - 1);
      out_hidden[idx] = hl[b * HLS + k];
    }
  }
}

// ---------------------------------------------------------------------------
// Kernel 3: per-b projection GEMM  out[b, v, 1+u] = proj_w @ st_b^T + proj_b
// 64x64 C tile per WG (8 waves, 2 tiles each), LDS staged (k-paired B),
// float4 global staging, coalesced s-stores.
// ---------------------------------------------------------------------------
__global__ void proj_kernel(const float* __restrict__ proj_w,
                            const float* __restrict__ proj_b,
                            const float* __restrict__ st,
                            float* __restrict__ out) {
  __shared__ float As[64 * AS_S];
  __shared__ float Bs[(KBLK / 2) * PB];
  const int tid = threadIdx.x;
  const int v0 = blockIdx.x * 64;
  const int u0 = blockIdx.y * 64;
  const int b  = blockIdx.z;
  const int wave = tid >> 5, lane = tid & 31;
  const int row = lane & 15, hi = lane >> 4;
  const int mt = wave & 3;        // 4 M tiles
  const int nth = wave >> 2;      // 2 N tile pairs
  const int nc0 = (nth * 2) * 16 + row;
  const int nc1 = (nth * 2 + 1) * 16 + row;
  v8f acc0 = {}, acc1 = {};

  for (int kb = 0; kb < HH; kb += KBLK) {
    // A: 64 rows x 32 k of proj_w (float4 coalesced)
    for (int idx = tid; idx < 64 * KBLK / 4; idx += 256) {
      int m = idx >> 3, k4 = idx & 7;
      int v = v0 + m;
      float4 val = (v < VV) ? *(const float4*)&proj_w[(size_t)v * HH + kb + 4 * k4]
                            : float4{0.f, 0.f, 0.f, 0.f};
      *(float4*)&As[m * AS_S + 4 * k4] = val;
    }
    // B: Bs[k/2][n][2] = st[u0+n][b][kb+k]  (float4 coalesced reads)
    for (int idx = tid; idx < 64 * KBLK / 4; idx += 256) {
      int n = idx >> 3, k4 = idx & 7;
      int uu = u0 + n;
      float4 val = (uu < NSTEP)
                       ? *(const float4*)&st[(size_t)uu * BB * HH + b * HH + kb + 4 * k4]
                       : float4{0.f, 0.f, 0.f, 0.f};
      v2f lo = {val.x, val.y}, hiv = {val.z, val.w};
      *(v2f*)&Bs[(2 * k4) * PB + 2 * n]     = lo;
      *(v2f*)&Bs[(2 * k4 + 1) * PB + 2 * n] = hiv;
    }
    __syncthreads();
#pragma unroll
    for (int k0 = 0; k0 < KBLK; k0 += 4) {
      v2f a  = *(const v2f*)&As[(mt * 16 + row) * AS_S + k0 + 2 * hi];
      v2f b0 = *(const v2f*)&Bs[((k0 >> 1) + hi) * PB + 2 * nc0];
      v2f b1 = *(const v2f*)&Bs[((k0 >> 1) + hi) * PB + 2 * nc1];
      acc0 = __builtin_amdgcn_wmma_f32_16x16x4_f32(false, a, false, b0,
                                                   (short)0, acc0, false, false);
      acc1 = __builtin_amdgcn_wmma_f32_16x16x4_f32(false, a, false, b1,
                                                   (short)0, acc1, false, false);
    }
    __syncthreads();
  }

#pragma unroll
  for (int r = 0; r < 8; ++r) {
    int v = v0 + mt * 16 + hi * 8 + r;
    if (v < VV) {
      float pb = proj_b[v];
      int ua = u0 + (nth * 2) * 16 + row;
      int ub = u0 + (nth * 2 + 1) * 16 + row;
      size_t base = ((size_t)b * VV + v) * SDIM + 1;
      if (ua < NSTEP) out[base + ua] = acc0[r] + pb;   // coalesced along s
      if (ub < NSTEP) out[base + ub] = acc1[r] + pb;
    }
  }
}

// ---------------------------------------------------------------------------
extern "C" void kernel_launch(void* const* d_in, const int* in_sizes, int n_in,
                              void* d_out, int out_size, void* d_ws, size_t ws_size,
                              hipStream_t stream) {
  const float* img_feat = (const float*)d_in[0];
  const int*   captions = (const int*)d_in[1];
  const float* embed    = (const float*)d_in[2];
  const float* w_ih     = (const float*)d_in[3];
  const float* w_hh     = (const float*)d_in[4];
  const float* b_ih     = (const float*)d_in[5];
  const float* b_hh     = (const float*)d_in[6];
  const float* proj_w   = (const float*)d_in[7];
  const float* proj_b   = (const float*)d_in[8];
  float* out = (float*)d_out;

  // Workspace layout (floats): Gi | hbuf | st | sync counter  (~53.4 MB)
  float* Gi   = (float*)d_ws;
  float* hbuf = Gi + (size_t)(TT - 1) * BB * G3H;
  float* st   = hbuf + BB * HH;
  unsigned* syncp = (unsigned*)(st + (size_t)NSTEP * BB * HH);

  init_kernel<<<(BB * VV + 255) / 256, 256, 0, stream>>>(img_feat, hbuf, out, syncp);
  gi_kernel<<<dim3(G3H / 256, BB, TT - 1), 256, 0, stream>>>(captions, embed, w_ih, b_ih, Gi);
  size_t lds = (size_t)(BB * HLS + (HH / 2) * PSTR + BB * GCOLS) * sizeof(float);
  gru_kernel<<<NWG, 384, lds, stream>>>(Gi, w_hh, b_hh, hbuf, st, syncp,
                                        out + (size_t)BB * VV * SDIM);
  proj_kernel<<<dim3((VV + 63) / 64, (NSTEP + 63) / 64, BB), 256, 0, stream>>>(
      proj_w, proj_b, st, out);
}